// FFT_Conv_Layer_88536455840524
// MI455X (gfx1250) — compile-verified
//
#include <hip/hip_runtime.h>

typedef __attribute__((ext_vector_type(2))) float v2f;
typedef __attribute__((ext_vector_type(8))) float v8f;

// Problem constants
// imgs:  [4, 32, 64, 64] f32
// filts1:[1, 32, 32, 3, 3, 2]  (inC, outC, fy, fx, re/im)
// filts2:[1, 32, 64, 3, 3, 2]
// filts3:[1, 64, 64, 3, 3, 2]
// out:   [4, 64, 64, 64] f32
//
// ws layout:
//   A5 (float2): [32][64][5][5] complex  @ offset 0        (409,600 B)
//   WT (float) : [64][1568]              @ offset 512*1024 (401,408 B)

#define ISTR 34  // padded channel stride in LDS (32 + 2): conflict-free, b64-aligned

// ---- Stage A: A5[i][c2] = sum_c1 h1[i,c1] (*) h2[c1,c2]   (complex 5x5) ----
__global__ void combine12_kernel(const float* __restrict__ f1,
                                 const float* __restrict__ f2,
                                 float2* __restrict__ A5) {
  int tid = blockIdx.x * blockDim.x + threadIdx.x;
  if (tid >= 32 * 64 * 25) return;
  int i   = tid / (64 * 25);
  int rem = tid % (64 * 25);
  int c2  = rem / 25;
  int t   = rem % 25;
  int ay = t / 5, ax = t % 5;
  float sr = 0.f, si = 0.f;
  for (int c1 = 0; c1 < 32; ++c1) {
    for (int uy = 0; uy < 3; ++uy) {
      int vy = ay - uy; if (vy < 0 || vy > 2) continue;
      for (int ux = 0; ux < 3; ++ux) {
        int vx = ax - ux; if (vx < 0 || vx > 2) continue;
        const float* p1 = f1 + (((i  * 32 + c1) * 3 + uy) * 3 + ux) * 2;
        const float* p2 = f2 + (((c1 * 64 + c2) * 3 + vy) * 3 + vx) * 2;
        float r1 = p1[0], i1 = p1[1], r2 = p2[0], i2 = p2[1];
        sr += r1 * r2 - i1 * i2;
        si += r1 * i2 + i1 * r2;
      }
    }
  }
  A5[tid] = make_float2(sr, si);
}

// ---- Stage B: WT[o][tap*32+i] = Re( sum_c2 A5[i,c2] (*) h3[c2,o] )[ky][kx] ----
__global__ void combine3_kernel(const float2* __restrict__ A5,
                                const float* __restrict__ f3,
                                float* __restrict__ WT) {
  int tid = blockIdx.x * blockDim.x + threadIdx.x;
  if (tid >= 32 * 64 * 49) return;
  int i   = tid / (64 * 49);
  int rem = tid % (64 * 49);
  int o   = rem / 49;
  int t   = rem % 49;       // tap = ky*7 + kx
  int ky = t / 7, kx = t % 7;
  float acc = 0.f;
  for (int c2 = 0; c2 < 64; ++c2) {
    const float2* Ab = A5 + (i  * 64 + c2) * 25;
    const float*  Fb = f3 + (c2 * 64 + o ) * 18;
    for (int vy = 0; vy < 3; ++vy) {
      int uy = ky - vy; if (uy < 0 || uy > 4) continue;
      for (int vx = 0; vx < 3; ++vx) {
        int ux = kx - vx; if (ux < 0 || ux > 4) continue;
        float2 a = Ab[uy * 5 + ux];
        float hr = Fb[(vy * 3 + vx) * 2 + 0];
        float hi = Fb[(vy * 3 + vx) * 2 + 1];
        acc += a.x * hr - a.y * hi;   // real part only (image is real)
      }
    }
  }
  WT[o * 1568 + t * 32 + i] = acc;
}

// ---- Main: out[b,o,y,x] = sum_{i,ky,kx} WT * img_circ(b,i,y+1-ky,x+1-kx) ----
// LDS layout: ext[ky][j][i]  (i innermost, stride ISTR=34)
//   j - 5 = circular column offset over the 66-domain; 7*70*34 floats = 66,640 B (dynamic)
__global__ __launch_bounds__(256) void fftconv_main_kernel(
    const float* __restrict__ img,
    const float* __restrict__ WT,
    float* __restrict__ out) {
  extern __shared__ float ext[];

  const int b = blockIdx.x >> 6;
  const int y = blockIdx.x & 63;
  const int tid = threadIdx.x;

  // Fill LDS: ext[(ky*70 + j)*ISTR + i] = imgp[(y+1-ky) mod 66][(j-5) mod 66]
  for (int n = tid; n < 32 * 7 * 70; n += 256) {
    int j   = n % 70;
    int row = n / 70;       // i*7 + ky
    int ky  = row % 7;
    int i   = row / 7;
    int r = y + 1 - ky; if (r < 0) r += 66;
    int p = j - 5;      if (p < 0) p += 66;
    float v = 0.f;
    if (r < 64 && p < 64)
      v = img[((b * 32 + i) * 64 + r) * 64 + p];
    ext[(ky * 70 + j) * ISTR + i] = v;
  }
  __syncthreads();

  const int lane = tid & 31;
  const int w    = tid >> 5;       // 8 waves
  const int m    = lane & 15;      // M / N index within tile
  const int hi   = lane >> 4;      // half-wave selector (K split)
  const int o_tile = (w & 3) * 16; // 4 o-tiles
  const int xp     = (w >> 2) * 32;
  const int xb0 = xp, xb1 = xp + 16;
  const int o = o_tile + m;

  v8f acc0 = {};
  v8f acc1 = {};

  // A-fragment base: WT[o][k], lane holds K = k0 + 2*hi, k0 + 2*hi + 1 (b64 aligned)
  const float* wtb = WT + o * 1568 + 2 * hi;

  for (int ky = 0; ky < 7; ++ky) {
    for (int kx = 0; kx < 7; ++kx) {
      const int tap = ky * 7 + kx;
      // B-fragment base for this lane: column j = (xb + m) + 6 - kx, channels 2*hi..
      const float* pb0 = &ext[(ky * 70 + (m + 6 - kx) + xb0) * ISTR + 2 * hi];
      const float* pb1 = pb0 + 16 * ISTR;
      const float* wk  = wtb + tap * 32;
#pragma unroll
      for (int s = 0; s < 8; ++s) {
        v2f a  = *(const v2f*)&wk[4 * s];   // K = tap*32 + 4s + {2hi, 2hi+1}
        v2f b0 = *(const v2f*)&pb0[4 * s];  // i = 4s + {2hi, 2hi+1}, adjacent in LDS
        v2f b1 = *(const v2f*)&pb1[4 * s];
        acc0 = __builtin_amdgcn_wmma_f32_16x16x4_f32(
            false, a, false, b0, (short)0, acc0, false, false);
        acc1 = __builtin_amdgcn_wmma_f32_16x16x4_f32(
            false, a, false, b1, (short)0, acc1, false, false);
      }
    }
  }

  // D layout: VGPR v -> M = v + 8*hi, N = m
#pragma unroll
  for (int v = 0; v < 8; ++v) {
    const int oo = o_tile + v + 8 * hi;
    out[((b * 64 + oo) * 64 + y) * 64 + xb0 + m] = acc0[v];
    out[((b * 64 + oo) * 64 + y) * 64 + xb1 + m] = acc1[v];
  }
}

extern "C" void kernel_launch(void* const* d_in, const int* in_sizes, int n_in,
                              void* d_out, int out_size, void* d_ws, size_t ws_size,
                              hipStream_t stream) {
  const float* imgs = (const float*)d_in[0];
  const float* f1   = (const float*)d_in[1];
  const float* f2   = (const float*)d_in[2];
  const float* f3   = (const float*)d_in[3];
  float* out = (float*)d_out;

  float2* A5 = (float2*)d_ws;
  float*  WT = (float*)((char*)d_ws + 512 * 1024);

  // Stage A: 32*64*25 = 51200 threads
  combine12_kernel<<<dim3(200), dim3(256), 0, stream>>>(f1, f2, A5);
  // Stage B: 32*64*49 = 100352 threads
  combine3_kernel<<<dim3(392), dim3(256), 0, stream>>>(A5, f3, WT);
  // Main: one WG per (b, y) row; dynamic LDS = 7*70*34 floats
  fftconv_main_kernel<<<dim3(4 * 64), dim3(256), 7 * 70 * ISTR * 4, stream>>>(imgs, WT, out);
}